// DynamicHybridRouter_39702677684789
// MI455X (gfx1250) — compile-verified
//
#include <hip/hip_runtime.h>
#include <hip/hip_bf16.h>
#include <math.h>

typedef __attribute__((ext_vector_type(2))) float v2f;
typedef __attribute__((ext_vector_type(4))) float v4f;
typedef __attribute__((ext_vector_type(8))) float v8f;

#define KDIM      2048
#define EXPERTS   64
#define ROWS_PER_WG 16
#define TEMPERATURE 2.0f

// One 128-thread WG (4 wave32) computes a 16-row x 64-expert logits tile with
// V_WMMA_F32_16X16X4_F32, then fuses bias + top-2 + softmax + scatter.
//
// WMMA f32 16x16x4 hardware operand mapping (ISA 7.12.2, wave32):
//   A (16x4, MxK): lane l -> M = l%16 ; vgpr v, half h=l/16 -> K-slot = 2h + v
//   B (4x16, KxN): lane l -> N = l%16 ; vgpr v, half h=l/16 -> K-slot = 2h + v
//   C/D (16x16):   lane l -> N = l%16 ; vgpr v, half h=l/16 -> M = 8h + v
//
// K-slot -> absolute-k remap (per 16-K chunk, A and B agree, each k summed once):
//   WMMA step i (0..3), half h, vgpr v  <->  k = k0 + 8h + 2i + v
// => each lane consumes k0+8h+{0..7}: two contiguous b128 loads per operand.
__global__ __launch_bounds__(128, 4)
void DynamicHybridRouter_kernel(const float* __restrict__ x,
                                const float* __restrict__ gate_w,
                                const float* __restrict__ gate_b,
                                const int*   __restrict__ maturity,
                                float* __restrict__ out)
{
    __shared__ float s_logits[ROWS_PER_WG][EXPERTS + 1];  // +1 pad: 65-bank stride
    __shared__ float s_p0[ROWS_PER_WG], s_p1[ROWS_PER_WG];
    __shared__ int   s_i0[ROWS_PER_WG], s_i1[ROWS_PER_WG];
    __shared__ int   s_immature;

    const int tid  = threadIdx.x;
    const int wave = tid >> 5;          // 0..3 -> expert block
    const int lane = tid & 31;
    const int m    = lane & 15;         // M for A-tile, N(expert) for B-tile
    const int h    = lane >> 4;         // lane-half selector
    const int row_base = blockIdx.x * ROWS_PER_WG;
    const int e_base   = wave * 16;

    if (tid == 0) s_immature = 0;
    __syncthreads();
    if (tid < EXPERTS) {
        if (maturity[tid] == 0) s_immature = 1;   // benign race, all write 1
    }

    // Per-lane operand base pointers (per-chunk +8h folded in once).
    const float* a_ptr = x      + (size_t)(row_base + m) * KDIM + 8 * h;
    const float* b_ptr = gate_w + (size_t)(e_base   + m) * KDIM + 8 * h;

    v8f acc = {0.f, 0.f, 0.f, 0.f, 0.f, 0.f, 0.f, 0.f};

#pragma unroll 2
    for (int k0 = 0; k0 < KDIM; k0 += 16) {
        // 4 x global_load_b128 per chunk; all issued before the WMMAs so
        // s_wait_loadcnt staging overlaps load return with matrix math.
        v4f aL = *(const v4f*)(a_ptr + k0);
        v4f bL = *(const v4f*)(b_ptr + k0);
        v4f aH = *(const v4f*)(a_ptr + k0 + 4);
        v4f bH = *(const v4f*)(b_ptr + k0 + 4);

        // Prefetch the x stream ~256B ahead (first-touch HBM latency cover;
        // speculative OOB prefetch at the tail is dropped by hardware).
        __builtin_prefetch(a_ptr + k0 + 64, 0, 3);

        acc = __builtin_amdgcn_wmma_f32_16x16x4_f32(
            false, __builtin_shufflevector(aL, aL, 0, 1),
            false, __builtin_shufflevector(bL, bL, 0, 1), (short)0, acc, false, false);
        acc = __builtin_amdgcn_wmma_f32_16x16x4_f32(
            false, __builtin_shufflevector(aL, aL, 2, 3),
            false, __builtin_shufflevector(bL, bL, 2, 3), (short)0, acc, false, false);
        acc = __builtin_amdgcn_wmma_f32_16x16x4_f32(
            false, __builtin_shufflevector(aH, aH, 0, 1),
            false, __builtin_shufflevector(bH, bH, 0, 1), (short)0, acc, false, false);
        acc = __builtin_amdgcn_wmma_f32_16x16x4_f32(
            false, __builtin_shufflevector(aH, aH, 2, 3),
            false, __builtin_shufflevector(bH, bH, 2, 3), (short)0, acc, false, false);
    }

    // Bias add + spill tile to LDS. D element (v,lane): row = 8h+v, expert = e_base+m.
    const float bias = gate_b[e_base + m];
#pragma unroll
    for (int v = 0; v < 8; ++v) {
        s_logits[8 * h + v][e_base + m] = acc[v] + bias;
    }
    __syncthreads();

    const bool immature = (s_immature != 0);

    if (!immature) {
        // --- top-2 branch (matches jax.lax.top_k: ties -> lowest index) ---
        if (tid < ROWS_PER_WG) {
            float v0 = -INFINITY, v1 = -INFINITY;
            int   i0 = 0,         i1 = 0;
#pragma unroll 4
            for (int e = 0; e < EXPERTS; ++e) {
                float lg = s_logits[tid][e];
                if (lg > v0) { v1 = v0; i1 = i0; v0 = lg; i0 = e; }
                else if (lg > v1) { v1 = lg; i1 = e; }
            }
            // softmax over [v0, v1]; v0 >= v1 so shift by v0.
            float e1  = __expf(v1 - v0);
            float inv = 1.0f / (1.0f + e1);
            s_p0[tid] = inv;        s_p1[tid] = e1 * inv;
            s_i0[tid] = i0;         s_i1[tid] = i1;
        }
        __syncthreads();
        // Scatter: 128 threads x 8 contiguous floats = 16x64 tile.
        const int r     = tid >> 3;
        const int cbase = (tid & 7) * 8;
        const int i0 = s_i0[r], i1 = s_i1[r];
        const float p0 = s_p0[r], p1 = s_p1[r];
        float* orow = out + (size_t)(row_base + r) * EXPERTS;
#pragma unroll
        for (int j = 0; j < 8; ++j) {
            int e = cbase + j;
            orow[e] = (e == i0) ? p0 : ((e == i1) ? p1 : 0.0f);
        }
    } else {
        // --- temperature-softmax branch (cold for these inputs) ---
        if (tid < ROWS_PER_WG) {
            float mx = -INFINITY;
            for (int e = 0; e < EXPERTS; ++e)
                mx = fmaxf(mx, s_logits[tid][e]);
            float sum = 0.0f;
            float* orow = out + (size_t)(row_base + tid) * EXPERTS;
            for (int e = 0; e < EXPERTS; ++e) {
                float p = __expf((s_logits[tid][e] - mx) * (1.0f / TEMPERATURE));
                orow[e] = p;
                sum += p;
            }
            float inv = 1.0f / sum;
            for (int e = 0; e < EXPERTS; ++e)
                orow[e] *= inv;
        }
    }
}

extern "C" void kernel_launch(void* const* d_in, const int* in_sizes, int n_in,
                              void* d_out, int out_size, void* d_ws, size_t ws_size,
                              hipStream_t stream) {
    const float* x    = (const float*)d_in[0];
    const float* gw   = (const float*)d_in[1];
    const float* gb   = (const float*)d_in[2];
    const int*   mat  = (const int*)d_in[3];
    float*       out  = (float*)d_out;

    const int n_rows = in_sizes[0] / KDIM;          // 16384
    const int n_wgs  = n_rows / ROWS_PER_WG;        // 1024

    DynamicHybridRouter_kernel<<<n_wgs, 128, 0, stream>>>(x, gw, gb, mat, out);
}